// TransformerUpdate_13932873909294
// MI455X (gfx1250) — compile-verified
//
#include <hip/hip_runtime.h>
#include <math.h>

// ---------------------------------------------------------------------------
// MI455X / gfx1250 implementation of the equivariant transformer update.
//
// Math notes:
//  * NormActivation(relu) is the exact identity (n>0 -> relu(n)/n == 1.0f),
//    so it is dropped.
//  * tp_dot(q[dst], k) == dot(qw[dst], k) with qw = per-l (q @ Wd) * tp_norm,
//    so Wd is folded into a per-node transform BEFORE the gather.
//  * edge i uses k[i], v[i] with i == node index, so k and v are computed
//    in-tile per 16-node slab and never hit HBM.
//
// CDNA5 specifics used:
//  * V_WMMA_F32_16X16X4_F32 for all per-l multiplicity mixes (fp32 GEMMs).
//  * TENSOR_LOAD_TO_LDS (TDM) to DMA the 32x152 feature slab global->LDS,
//    bypassing VGPRs; completion via s_wait_tensorcnt.
//  * global_prefetch for the random qw[dst] gather (240MB working set > L2).
//  * wave32 shfl reduction, native global f32 atomic fadd scatters.
// ---------------------------------------------------------------------------

typedef float v2f __attribute__((ext_vector_type(2)));
typedef float v8f __attribute__((ext_vector_type(8)));
typedef unsigned int u32x4 __attribute__((ext_vector_type(4)));
typedef int i32x4 __attribute__((ext_vector_type(4)));
typedef int i32x8 __attribute__((ext_vector_type(8)));

#define F_IN   152     // 64*1 + 16*3 + 8*5
#define F_QK   120     // 32*1 + 16*3 + 8*5
#define TILE   16
#define WPB    2       // 16-node tiles (waves) per block
#define NTHREADS (WPB * 32)

// per-l 1/sqrt(mul_in) normalizations
#define SC_L0  0.125f                  // 1/sqrt(64)
#define SC_L1  0.25f                   // 1/sqrt(16)
#define SC_L2  0.35355339059327373f    // 1/sqrt(8)
// tp_dot per-l norms folded into qw: 1/sqrt(mul*mul*d)
#define TP_L0  0.03125f                // 1/sqrt(32*32*1)
#define TP_L1  0.036084391824351615f   // 1/sqrt(16*16*3)
#define TP_L2  0.05590169943749474f    // 1/sqrt(8*8*5)

// D(16x16) += A(16xK) * B(Kx16), A/B read from LDS, fp32 WMMA.
// A fragment (ISA 7.12.2): lanes 0-15 hold M=0..15; VGPR0 = K, VGPR1 = K+1;
// half-wave 1 covers K+2/K+3. B mirrors A with K as the row dimension.
__device__ __forceinline__ v8f wmma_gemm(const float* A, int ars, int aks,
                                         const float* B, int brs, int K,
                                         v8f c, int lane) {
  const int mn = lane & 15;
  const int kb = (lane >> 4) << 1;   // 0 or 2
  #pragma unroll
  for (int k = 0; k < K; k += 4) {
    v2f a, b;
    a.x = A[mn * ars + (k + kb) * aks];
    a.y = A[mn * ars + (k + kb + 1) * aks];
    b.x = B[(k + kb) * brs + mn];
    b.y = B[(k + kb + 1) * brs + mn];
    c = __builtin_amdgcn_wmma_f32_16x16x4_f32(false, a, false, b,
                                              (short)0, c, false, false);
  }
  return c;
}

// C/D fragment (ISA 7.12.2): VGPR r holds row r (lanes 0-15) / row r+8
// (lanes 16-31), column = lane&15.  Scatter with row/col strides + scale.
__device__ __forceinline__ void store_tile(float* dst, int rs, int cs,
                                           v8f c, float scale, int lane,
                                           int ncols) {
  const int n = lane & 15;
  const int half = lane >> 4;
  if (n < ncols) {
    #pragma unroll
    for (int r = 0; r < 8; ++r)
      dst[(r + 8 * half) * rs + n * cs] = c[r] * scale;
  }
}

// Cooperative weight load; pads missing columns with zeros (8-wide l2 mats
// are stored 8x16 so the 16-wide WMMA B fragment reads zeros past col 7).
__device__ __forceinline__ void load_w(float* dst, const float* src,
                                       int rows, int cols, int dld,
                                       int tid, int nt) {
  for (int i = tid; i < rows * dld; i += nt) {
    int r = i / dld, cc = i % dld;
    dst[i] = (cc < cols) ? src[r * cols + cc] : 0.0f;
  }
}

__device__ __forceinline__ void atomic_add_f32(float* p, float v) {
  __hip_atomic_fetch_add(p, v, __ATOMIC_RELAXED, __HIP_MEMORY_SCOPE_AGENT);
}

// ---------------------------------------------------------------------------
// TDM: DMA a [rows x 152] f32 slab (row stride 152) from global to LDS.
// Descriptor packed per CDNA5 ISA 8.3/8.4 (D# group0 128b, group1 256b;
// groups 2/3 + trailing group zero => 2D tensor).  tensor_dim1 = rows_valid
// so tail rows beyond N read back as zero (TDM OOB semantics).  One call per
// block, issued by wave 0 only; TENSORcnt is per-wave so only wave 0 waits,
// then __syncthreads() publishes the tile.
// ---------------------------------------------------------------------------
__device__ __forceinline__ void tdm_load_x(const float* gsrc, float* ldsdst,
                                           int rows_valid) {
  const unsigned long long ga = (unsigned long long)(uintptr_t)gsrc;
  const unsigned int lds = (unsigned int)(uintptr_t)ldsdst;  // low 32b = LDS offset
  u32x4 g0;
  g0[0] = 1u;                                   // count=1, no gather/iterate
  g0[1] = lds;                                  // lds_addr
  g0[2] = (unsigned int)ga;                     // global_addr[31:0]
  g0[3] = (unsigned int)((ga >> 32) & 0x1FFFFFFu)  // global_addr[56:32]
          | 0x80000000u;                        // type=2 ("image")
  i32x8 g1;
  const unsigned int rv = (unsigned int)rows_valid;
  g1[0] = (int)(2u << 16);                      // data_size = 4 bytes
  g1[1] = (int)(152u << 16);                    // tensor_dim0[15:0] = 152
  g1[2] = (int)((rv & 0xFFFFu) << 16);          // tensor_dim1[15:0] = rows
  g1[3] = (int)(((rv >> 16) & 0xFFFFu)          // tensor_dim1[31:16]
          | (152u << 16));                      // tile_dim0 = 152
  g1[4] = (int)(WPB * TILE);                    // tile_dim1 = 32, tile_dim2=0
  g1[5] = 152;                                  // tensor_dim0_stride[31:0]
  g1[6] = 0;                                    // stride hi / dim1_stride lo
  g1[7] = 0;
  const i32x4 z4 = {0, 0, 0, 0};
  const i32x8 z8 = {0, 0, 0, 0, 0, 0, 0, 0};
  // 6-arg form on this toolchain: (g0, g1, g2, g3, extra, cpol)
  __builtin_amdgcn_tensor_load_to_lds(g0, g1, z4, z4, z8, 0);
}

// ---- kernel 0: zero destination + softmax denominators ----
__global__ void k_zero(float* __restrict__ out, float* __restrict__ z,
                       long nout, long nz) {
  long i = (long)blockIdx.x * blockDim.x + threadIdx.x;
  long stride = (long)gridDim.x * blockDim.x;
  for (long j = i; j < nout; j += stride) out[j] = 0.0f;
  for (long j = i; j < nz; j += stride) z[j] = 0.0f;
}

// ---- kernel 1: qw[n] = tp_norm * Wd-mix( q-mix(x[n]) ) ----
__global__ __launch_bounds__(NTHREADS) void k_qw(
    const float* __restrict__ X,
    const float* __restrict__ Wq0, const float* __restrict__ Wq1,
    const float* __restrict__ Wq2,
    const float* __restrict__ Wd0, const float* __restrict__ Wd1,
    const float* __restrict__ Wd2,
    float* __restrict__ qw, int ntiles, int N) {
  __shared__ float sWq0[64 * 32], sWq1[16 * 16], sWq2[8 * 16];
  __shared__ float sWd0[32 * 32], sWd1[16 * 16], sWd2[8 * 16];
  __shared__ float Xt[WPB][TILE * F_IN];
  __shared__ float Qt[WPB][TILE * F_QK];

  const int tid = threadIdx.x, wave = tid >> 5, lane = tid & 31;

  // async DMA of this block's 32x152 feature slab into LDS (wave 0 issues)
  if (wave == 0) {
    const long row0 = (long)blockIdx.x * WPB * TILE;
    long rem = (long)N - row0;
    const int rows = (int)(rem < (long)(WPB * TILE) ? (rem < 0 ? 0 : rem)
                                                    : (long)(WPB * TILE));
    tdm_load_x(X + row0 * F_IN, &Xt[0][0], rows);
  }

  load_w(sWq0, Wq0, 64, 32, 32, tid, NTHREADS);
  load_w(sWq1, Wq1, 16, 16, 16, tid, NTHREADS);
  load_w(sWq2, Wq2, 8, 8, 16, tid, NTHREADS);
  load_w(sWd0, Wd0, 32, 32, 32, tid, NTHREADS);
  load_w(sWd1, Wd1, 16, 16, 16, tid, NTHREADS);
  load_w(sWd2, Wd2, 8, 8, 16, tid, NTHREADS);

  if (wave == 0) __builtin_amdgcn_s_wait_tensorcnt(0);
  __syncthreads();

  const int tile = blockIdx.x * WPB + wave;
  if (tile >= ntiles) return;   // grid is sized exactly; never taken

  float* xt = Xt[wave];
  float* qt = Qt[wave];

  // q = o3_linear(x, Wq)   (norm_act == identity)
  for (int co = 0; co < 32; co += 16) {
    v8f c = {};
    c = wmma_gemm(xt, F_IN, 1, sWq0 + co, 32, 64, c, lane);
    store_tile(qt + co, F_QK, 1, c, SC_L0, lane, 16);
  }
  for (int d = 0; d < 3; ++d) {
    v8f c = {};
    c = wmma_gemm(xt + 64 + d, F_IN, 3, sWq1, 16, 16, c, lane);
    store_tile(qt + 32 + d, F_QK, 3, c, SC_L1, lane, 16);
  }
  for (int d = 0; d < 5; ++d) {
    v8f c = {};
    c = wmma_gemm(xt + 112 + d, F_IN, 5, sWq2, 16, 8, c, lane);
    store_tile(qt + 80 + d, F_QK, 5, c, SC_L2, lane, 8);
  }
  __syncthreads();

  // qw = (q @ Wd) * tp_norm  -> global scratch (gathered later by dst)
  float* qwg = qw + (long)tile * TILE * F_QK;
  for (int co = 0; co < 32; co += 16) {
    v8f c = {};
    c = wmma_gemm(qt, F_QK, 1, sWd0 + co, 32, 32, c, lane);
    store_tile(qwg + co, F_QK, 1, c, TP_L0, lane, 16);
  }
  for (int d = 0; d < 3; ++d) {
    v8f c = {};
    c = wmma_gemm(qt + 32 + d, F_QK, 3, sWd1, 16, 16, c, lane);
    store_tile(qwg + 32 + d, F_QK, 3, c, TP_L1, lane, 16);
  }
  for (int d = 0; d < 5; ++d) {
    v8f c = {};
    c = wmma_gemm(qt + 80 + d, F_QK, 5, sWd2, 16, 8, c, lane);
    store_tile(qwg + 80 + d, F_QK, 5, c, TP_L2, lane, 8);
  }
}

// ---- kernel 2: e[i] = exp(dot(qw[dst[i]], k[i])), z[dst] += e ----
__global__ __launch_bounds__(NTHREADS) void k_score(
    const float* __restrict__ X,
    const float* __restrict__ Wk0, const float* __restrict__ Wk1,
    const float* __restrict__ Wk2,
    const int* __restrict__ edge_dst,
    const float* __restrict__ qw,
    float* __restrict__ e_ws, float* __restrict__ z,
    int ntiles, int N) {
  __shared__ float sW0[64 * 32], sW1[16 * 16], sW2[8 * 16];
  __shared__ float Xt[WPB][TILE * F_IN];
  __shared__ float Kt[WPB][TILE * F_QK];

  const int tid = threadIdx.x, wave = tid >> 5, lane = tid & 31;

  if (wave == 0) {
    const long row0 = (long)blockIdx.x * WPB * TILE;
    long rem = (long)N - row0;
    const int rows = (int)(rem < (long)(WPB * TILE) ? (rem < 0 ? 0 : rem)
                                                    : (long)(WPB * TILE));
    tdm_load_x(X + row0 * F_IN, &Xt[0][0], rows);
  }

  load_w(sW0, Wk0, 64, 32, 32, tid, NTHREADS);
  load_w(sW1, Wk1, 16, 16, 16, tid, NTHREADS);
  load_w(sW2, Wk2, 8, 8, 16, tid, NTHREADS);

  if (wave == 0) __builtin_amdgcn_s_wait_tensorcnt(0);
  __syncthreads();

  const int tile = blockIdx.x * WPB + wave;
  if (tile >= ntiles) return;

  float* xt = Xt[wave];
  float* kt = Kt[wave];

  // k = o3_linear(x, Wk)  (in-tile, never stored to HBM)
  for (int co = 0; co < 32; co += 16) {
    v8f c = {};
    c = wmma_gemm(xt, F_IN, 1, sW0 + co, 32, 64, c, lane);
    store_tile(kt + co, F_QK, 1, c, SC_L0, lane, 16);
  }
  for (int d = 0; d < 3; ++d) {
    v8f c = {};
    c = wmma_gemm(xt + 64 + d, F_IN, 3, sW1, 16, 16, c, lane);
    store_tile(kt + 32 + d, F_QK, 3, c, SC_L1, lane, 16);
  }
  for (int d = 0; d < 5; ++d) {
    v8f c = {};
    c = wmma_gemm(xt + 112 + d, F_IN, 5, sW2, 16, 8, c, lane);
    store_tile(kt + 80 + d, F_QK, 5, c, SC_L2, lane, 8);
  }
  __syncthreads();

  // one wave per edge-dot: lanes stride the 120 components, shfl-reduce.
  for (int e = 0; e < TILE; ++e) {
    const int ei = tile * TILE + e;
    if (ei >= N) break;
    const int dst = edge_dst[ei];
    // prefetch next edge's qw row while this one reduces (random gather,
    // 240MB working set > 192MB L2) -> global_prefetch
    if (e + 1 < TILE && ei + 1 < N)
      __builtin_prefetch(qw + (long)edge_dst[ei + 1] * F_QK, 0, 0);
    const float* qrow = qw + (long)dst * F_QK;   // coalesced gather of 1 row
    const float* krow = kt + e * F_QK;
    float p = 0.0f;
    for (int c2 = lane; c2 < F_QK; c2 += 32) p += qrow[c2] * krow[c2];
    #pragma unroll
    for (int off = 16; off; off >>= 1) p += __shfl_xor(p, off, 32);
    if (lane == 0) {
      const float ex = __expf(p);
      e_ws[ei] = ex;
      atomic_add_f32(&z[dst], ex);
    }
  }
}

// ---- kernel 3: out[dst] += sqrt(relu(e/z[dst])) * v[i] ----
__global__ __launch_bounds__(NTHREADS) void k_out(
    const float* __restrict__ X,
    const float* __restrict__ Wv0, const float* __restrict__ Wv1,
    const float* __restrict__ Wv2,
    const int* __restrict__ edge_dst,
    const float* __restrict__ e_ws, const float* __restrict__ z,
    float* __restrict__ out, int ntiles, int N) {
  __shared__ float sW0[64 * 64], sW1[16 * 16], sW2[8 * 16];
  __shared__ float Xt[WPB][TILE * F_IN];
  __shared__ float Vt[WPB][TILE * F_IN];

  const int tid = threadIdx.x, wave = tid >> 5, lane = tid & 31;

  if (wave == 0) {
    const long row0 = (long)blockIdx.x * WPB * TILE;
    long rem = (long)N - row0;
    const int rows = (int)(rem < (long)(WPB * TILE) ? (rem < 0 ? 0 : rem)
                                                    : (long)(WPB * TILE));
    tdm_load_x(X + row0 * F_IN, &Xt[0][0], rows);
  }

  load_w(sW0, Wv0, 64, 64, 64, tid, NTHREADS);
  load_w(sW1, Wv1, 16, 16, 16, tid, NTHREADS);
  load_w(sW2, Wv2, 8, 8, 16, tid, NTHREADS);

  if (wave == 0) __builtin_amdgcn_s_wait_tensorcnt(0);
  __syncthreads();

  const int tile = blockIdx.x * WPB + wave;
  if (tile >= ntiles) return;

  float* xt = Xt[wave];
  float* vt = Vt[wave];

  // v = o3_linear(x, Wv): 64->64 l0, 16->16 l1, 8->8 l2
  for (int co = 0; co < 64; co += 16) {
    v8f c = {};
    c = wmma_gemm(xt, F_IN, 1, sW0 + co, 64, 64, c, lane);
    store_tile(vt + co, F_IN, 1, c, SC_L0, lane, 16);
  }
  for (int d = 0; d < 3; ++d) {
    v8f c = {};
    c = wmma_gemm(xt + 64 + d, F_IN, 3, sW1, 16, 16, c, lane);
    store_tile(vt + 64 + d, F_IN, 3, c, SC_L1, lane, 16);
  }
  for (int d = 0; d < 5; ++d) {
    v8f c = {};
    c = wmma_gemm(xt + 112 + d, F_IN, 5, sW2, 16, 8, c, lane);
    store_tile(vt + 112 + d, F_IN, 5, c, SC_L2, lane, 8);
  }
  __syncthreads();

  for (int e = 0; e < TILE; ++e) {
    const int ei = tile * TILE + e;
    if (ei >= N) break;
    const int dst = edge_dst[ei];
    float zz = z[dst];
    zz = (zz > 0.0f) ? zz : 1.0f;
    float al = e_ws[ei] / zz;
    al = fmaxf(al, 0.0f);
    const float w = sqrtf(al);
    const float* vrow = vt + e * F_IN;
    float* orow = out + (long)dst * F_IN;
    for (int c2 = lane; c2 < F_IN; c2 += 32)
      atomic_add_f32(&orow[c2], w * vrow[c2]);   // coalesced fadd scatter
  }
}

extern "C" void kernel_launch(void* const* d_in, const int* in_sizes, int n_in,
                              void* d_out, int out_size, void* d_ws,
                              size_t ws_size, hipStream_t stream) {
  const float* X   = (const float*)d_in[0];
  const int* edst  = (const int*)d_in[1];
  const float* Wq0 = (const float*)d_in[2];
  const float* Wq1 = (const float*)d_in[3];
  const float* Wq2 = (const float*)d_in[4];
  const float* Wk0 = (const float*)d_in[5];
  const float* Wk1 = (const float*)d_in[6];
  const float* Wk2 = (const float*)d_in[7];
  const float* Wv0 = (const float*)d_in[8];
  const float* Wv1 = (const float*)d_in[9];
  const float* Wv2 = (const float*)d_in[10];
  const float* Wd0 = (const float*)d_in[11];
  const float* Wd1 = (const float*)d_in[12];
  const float* Wd2 = (const float*)d_in[13];
  float* out = (float*)d_out;

  const int N = in_sizes[1];                 // #edges == #nodes
  const int ntiles = (N + TILE - 1) / TILE;  // 31250 for N=500000

  // workspace: qw [ntiles*16 x 120] | e [ntiles*16] | z [N]
  float* qw = (float*)d_ws;
  float* e_ws = qw + (size_t)ntiles * TILE * F_QK;
  float* z = e_ws + (size_t)ntiles * TILE;

  k_zero<<<1024, 256, 0, stream>>>(out, z, (long)N * F_IN, (long)N);

  const int nblocks = (ntiles + WPB - 1) / WPB;
  k_qw<<<nblocks, NTHREADS, 0, stream>>>(X, Wq0, Wq1, Wq2, Wd0, Wd1, Wd2,
                                         qw, ntiles, N);
  k_score<<<nblocks, NTHREADS, 0, stream>>>(X, Wk0, Wk1, Wk2, edst, qw,
                                            e_ws, z, ntiles, N);
  k_out<<<nblocks, NTHREADS, 0, stream>>>(X, Wv0, Wv1, Wv2, edst, e_ws, z,
                                          out, ntiles, N);
}